// GraphSAGE_MUTAG_27633819582786
// MI455X (gfx1250) — compile-verified
//
#include <hip/hip_runtime.h>
#include <hip/hip_bf16.h>

typedef float v2f __attribute__((ext_vector_type(2)));
typedef float v8f __attribute__((ext_vector_type(8)));

// ---------------------------------------------------------------- utilities
__global__ void zero_f32_kernel(float* __restrict__ p, long long n) {
    long long i = (long long)blockIdx.x * blockDim.x + threadIdx.x;
    if (i < n) p[i] = 0.0f;
}

// pad x (N x 14) -> xp (N x 16), cols 14..15 = 0
__global__ void pack_x_kernel(const float* __restrict__ x, float* __restrict__ xp, int nNodes) {
    long long i = (long long)blockIdx.x * blockDim.x + threadIdx.x;
    long long total = (long long)nNodes * 16;
    if (i >= total) return;
    int node = (int)(i >> 4);
    int f = (int)(i & 15);
    xp[i] = (f < 14) ? x[(long long)node * 14 + f] : 0.0f;
}

// pad W (R x 64) -> Wp (RP x 64), rows >= R zeroed
__global__ void pack_w_rows_kernel(const float* __restrict__ W, float* __restrict__ Wp,
                                   int R, int RP) {
    int i = blockIdx.x * blockDim.x + threadIdx.x;
    if (i >= RP * 64) return;
    int r = i >> 6;
    Wp[i] = (r < R) ? W[i] : 0.0f;   // row-major, same 64-col stride
}

// pad W (64 x C) -> Wp (64 x 16), cols >= C zeroed
__global__ void pack_w_cols_kernel(const float* __restrict__ W, float* __restrict__ Wp, int C) {
    int i = blockIdx.x * blockDim.x + threadIdx.x;
    if (i >= 64 * 16) return;
    int r = i >> 4;
    int c = i & 15;
    Wp[i] = (c < C) ? W[r * C + c] : 0.0f;
}

// pad bias (C) -> bp (16)
__global__ void pack_b_kernel(const float* __restrict__ b, float* __restrict__ bp, int C) {
    int i = blockIdx.x * blockDim.x + threadIdx.x;
    if (i >= 16) return;
    bp[i] = (i < C) ? b[i] : 0.0f;
}

__global__ void degree_kernel(const long long* __restrict__ dst, float* __restrict__ deg, int nEdges) {
    int e = blockIdx.x * blockDim.x + threadIdx.x;
    if (e >= nEdges) return;
    int d = (int)dst[e];
    __hip_atomic_fetch_add(&deg[d], 1.0f, __ATOMIC_RELAXED, __HIP_MEMORY_SCOPE_AGENT);
}

__global__ void invdeg_kernel(const float* __restrict__ deg, float* __restrict__ invd, int n) {
    int i = blockIdx.x * blockDim.x + threadIdx.x;
    if (i < n) invd[i] = 1.0f / fmaxf(deg[i], 1.0f);
}

// -------------------------------------------------------- edge scatter-add
// STRIDE floats per node row, CH floats per thread chunk (STRIDE % CH == 0)
template <int STRIDE, int CH>
__global__ __launch_bounds__(256) void scatter_add_kernel(
    const long long* __restrict__ src, const long long* __restrict__ dst,
    const float* __restrict__ h, float* __restrict__ agg, int nEdges)
{
    constexpr int TPE = STRIDE / CH;  // threads per edge
    long long gid = (long long)blockIdx.x * blockDim.x + threadIdx.x;
    long long total = (long long)nEdges * TPE;
    if (gid >= total) return;
    int e  = (int)(gid / TPE);
    int f0 = (int)(gid % TPE) * CH;
    int s = (int)src[e];
    int d = (int)dst[e];
    const float* hp = h   + (long long)s * STRIDE + f0;
    float*       ap = agg + (long long)d * STRIDE + f0;
#pragma unroll
    for (int j = 0; j < CH; ++j) {
        __hip_atomic_fetch_add(ap + j, hp[j], __ATOMIC_RELAXED, __HIP_MEMORY_SCOPE_AGENT);
    }
}

// ------------------------------------------------------------- WMMA GEMM
// out[node, 0:OUTC] = act( (agg[node,:]*inv_deg[node]) @ Wl + xin[node,:] @ Wr + bias )
// Weights are pre-padded: Wl is K1PAD x WN, Wr is K2PAD x WN, WN = NTILES*16.
// One wave computes a 16-node x (NTILES*16)-col tile with V_WMMA_F32_16X16X4_F32.
// A layout: lane L -> row M = L%16, VGPR v holds K = k + 2*(L/16) + v.
// B layout: lane L -> col N = L%16, VGPR v holds K = k + 2*(L/16) + v.
// C/D layout: VGPR v, lanes0-15 -> M=v; lanes16-31 -> M=v+8; N = L%16.
template <int K1PAD, int K2PAD, int NTILES, int OUTC, bool RELU>
__global__ __launch_bounds__(256) void sage_gemm_kernel(
    const float* __restrict__ agg, int aggStride,
    const float* __restrict__ xin, int xStride,
    const float* __restrict__ invd,
    const float* __restrict__ Wl, const float* __restrict__ Wr,
    const float* __restrict__ bias,
    float* __restrict__ out, int outStride, int nTiles)
{
    constexpr int WN = NTILES * 16;
    const int wave = threadIdx.x >> 5;
    const int lane = threadIdx.x & 31;
    const int tile = blockIdx.x * 8 + wave;
    if (tile >= nTiles) return;  // wave-uniform: EXEC stays all-ones for WMMA

    const int half = lane >> 4;   // 0 or 1
    const int rc   = lane & 15;   // A-row / B-col within 16
    const int rowNode = tile * 16 + rc;
    const float scale = invd[rowNode];

    v8f acc[NTILES];
#pragma unroll
    for (int t = 0; t < NTILES; ++t) {
        v8f z = {0.f, 0.f, 0.f, 0.f, 0.f, 0.f, 0.f, 0.f};
        acc[t] = z;
    }

    const float* aRow = agg + (long long)rowNode * aggStride;
    const float* xRow = xin + (long long)rowNode * xStride;

    // ---- (agg * inv_deg) @ Wl
#pragma unroll
    for (int k = 0; k < K1PAD; k += 4) {
        const int ka = k + 2 * half;                       // even -> 8B aligned
        v2f a = *(const v2f*)(aRow + ka);
        a.x *= scale;
        a.y *= scale;
#pragma unroll
        for (int t = 0; t < NTILES; ++t) {
            const int col = t * 16 + rc;
            v2f b;
            b.x = Wl[ka * WN + col];
            b.y = Wl[(ka + 1) * WN + col];
            acc[t] = __builtin_amdgcn_wmma_f32_16x16x4_f32(
                false, a, false, b, (short)0, acc[t], false, false);
        }
    }

    // ---- xin @ Wr
#pragma unroll
    for (int k = 0; k < K2PAD; k += 4) {
        const int ka = k + 2 * half;
        v2f a = *(const v2f*)(xRow + ka);
#pragma unroll
        for (int t = 0; t < NTILES; ++t) {
            const int col = t * 16 + rc;
            v2f b;
            b.x = Wr[ka * WN + col];
            b.y = Wr[(ka + 1) * WN + col];
            acc[t] = __builtin_amdgcn_wmma_f32_16x16x4_f32(
                false, a, false, b, (short)0, acc[t], false, false);
        }
    }

    // ---- epilogue: bias (+ relu), write D (guard only the store columns)
#pragma unroll
    for (int t = 0; t < NTILES; ++t) {
        const int n = t * 16 + rc;
        const float bv = bias[n];
        if (n < OUTC) {
#pragma unroll
            for (int v = 0; v < 8; ++v) {
                const int m = v + 8 * half;
                float val = acc[t][v] + bv;
                if (RELU) val = fmaxf(val, 0.0f);
                out[(long long)(tile * 16 + m) * outStride + n] = val;
            }
        }
    }
}

// ------------------------------------------------------------- pooling
__global__ void pool_kernel(const float* __restrict__ h2,
                            const long long* __restrict__ batch,
                            float* __restrict__ gsum, float* __restrict__ gcnt, int n)
{
    int i = blockIdx.x * blockDim.x + threadIdx.x;
    if (i >= n) return;
    int g = (int)batch[i];
    __hip_atomic_fetch_add(&gsum[2 * g + 0], h2[2 * i + 0], __ATOMIC_RELAXED, __HIP_MEMORY_SCOPE_AGENT);
    __hip_atomic_fetch_add(&gsum[2 * g + 1], h2[2 * i + 1], __ATOMIC_RELAXED, __HIP_MEMORY_SCOPE_AGENT);
    __hip_atomic_fetch_add(&gcnt[g], 1.0f, __ATOMIC_RELAXED, __HIP_MEMORY_SCOPE_AGENT);
}

__global__ void final_kernel(const float* __restrict__ gsum, const float* __restrict__ gcnt,
                             float* __restrict__ out, int ng)
{
    int g = blockIdx.x * blockDim.x + threadIdx.x;
    if (g >= ng) return;
    float c  = fmaxf(gcnt[g], 1.0f);
    float p0 = gsum[2 * g + 0] / c;
    float p1 = gsum[2 * g + 1] / c;
    float m  = fmaxf(p0, p1);
    float lse = m + logf(expf(p0 - m) + expf(p1 - m));
    out[2 * g + 0] = p0 - lse;
    out[2 * g + 1] = p1 - lse;
}

// ------------------------------------------------------------- launcher
extern "C" void kernel_launch(void* const* d_in, const int* in_sizes, int n_in,
                              void* d_out, int out_size, void* d_ws, size_t ws_size,
                              hipStream_t stream)
{
    const float* x   = (const float*)d_in[0];
    const float* W1l = (const float*)d_in[1];
    const float* W1r = (const float*)d_in[2];
    const float* b1  = (const float*)d_in[3];
    const float* W2l = (const float*)d_in[4];
    const float* W2r = (const float*)d_in[5];
    const float* b2  = (const float*)d_in[6];
    const float* W3l = (const float*)d_in[7];
    const float* W3r = (const float*)d_in[8];
    const float* b3  = (const float*)d_in[9];
    const float* W4l = (const float*)d_in[10];
    const float* W4r = (const float*)d_in[11];
    const float* b4  = (const float*)d_in[12];
    const long long* edge_index = (const long long*)d_in[13];
    const long long* batch      = (const long long*)d_in[14];

    const int nNodes  = in_sizes[14];          // 100000
    const int nEdges  = in_sizes[13] / 2;      // 3200000
    const int nGraphs = out_size / 2;          // 2000
    const long long* src = edge_index;
    const long long* dst = edge_index + nEdges;

    float* out = (float*)d_out;

    // workspace layout (floats)
    float* ws    = (float*)d_ws;
    float* xp    = ws;                                   // nNodes*16
    float* hA    = xp   + (long long)nNodes * 16;        // nNodes*64
    float* hB    = hA   + (long long)nNodes * 64;        // nNodes*64
    float* agg   = hB   + (long long)nNodes * 64;        // nNodes*64
    float* deg   = agg  + (long long)nNodes * 64;        // nNodes
    float* invd  = deg  + nNodes;                        // nNodes
    float* gsum  = invd + nNodes;                        // nGraphs*2
    float* gcnt  = gsum + (long long)nGraphs * 2;        // nGraphs
    float* W1lp  = gcnt + nGraphs;                       // 16*64
    float* W1rp  = W1lp + 16 * 64;                       // 16*64
    float* W4lp  = W1rp + 16 * 64;                       // 64*16
    float* W4rp  = W4lp + 64 * 16;                       // 64*16
    float* b4p   = W4rp + 64 * 16;                       // 16
    float* h4    = hB;                                   // reuse: nNodes*2 (layer4 out)

    const int B = 256;
    auto blocks = [](long long n, int b) { return (int)((n + b - 1) / b); };
    const int nTiles = nNodes / 16;                      // 6250
    const int gemmBlocks = (nTiles + 7) / 8;

    // 1) zero deg/invd/gsum/gcnt (contiguous)
    {
        long long nz = (long long)nNodes * 2 + (long long)nGraphs * 3;
        zero_f32_kernel<<<blocks(nz, B), B, 0, stream>>>(deg, nz);
    }
    // 2) pad x and weights
    pack_x_kernel<<<blocks((long long)nNodes * 16, B), B, 0, stream>>>(x, xp, nNodes);
    pack_w_rows_kernel<<<blocks(16 * 64, B), B, 0, stream>>>(W1l, W1lp, 14, 16);
    pack_w_rows_kernel<<<blocks(16 * 64, B), B, 0, stream>>>(W1r, W1rp, 14, 16);
    pack_w_cols_kernel<<<blocks(64 * 16, B), B, 0, stream>>>(W4l, W4lp, 2);
    pack_w_cols_kernel<<<blocks(64 * 16, B), B, 0, stream>>>(W4r, W4rp, 2);
    pack_b_kernel<<<1, 16, 0, stream>>>(b4, b4p, 2);
    // 3) degree + inv_deg
    degree_kernel<<<blocks(nEdges, B), B, 0, stream>>>(dst, deg, nEdges);
    invdeg_kernel<<<blocks(nNodes, B), B, 0, stream>>>(deg, invd, nNodes);

    // ---- layer 1: xp (N x 16 padded, F_IN=14) -> hA (N x 64), relu
    zero_f32_kernel<<<blocks((long long)nNodes * 16, B), B, 0, stream>>>(agg, (long long)nNodes * 16);
    scatter_add_kernel<16, 4><<<blocks((long long)nEdges * 4, B), B, 0, stream>>>(src, dst, xp, agg, nEdges);
    sage_gemm_kernel<16, 16, 4, 64, true><<<gemmBlocks, B, 0, stream>>>(
        agg, 16, xp, 16, invd, W1lp, W1rp, b1, hA, 64, nTiles);

    // ---- layer 2: hA -> hB, relu
    zero_f32_kernel<<<blocks((long long)nNodes * 64, B), B, 0, stream>>>(agg, (long long)nNodes * 64);
    scatter_add_kernel<64, 8><<<blocks((long long)nEdges * 8, B), B, 0, stream>>>(src, dst, hA, agg, nEdges);
    sage_gemm_kernel<64, 64, 4, 64, true><<<gemmBlocks, B, 0, stream>>>(
        agg, 64, hA, 64, invd, W2l, W2r, b2, hB, 64, nTiles);

    // ---- layer 3: hB -> hA, relu
    zero_f32_kernel<<<blocks((long long)nNodes * 64, B), B, 0, stream>>>(agg, (long long)nNodes * 64);
    scatter_add_kernel<64, 8><<<blocks((long long)nEdges * 8, B), B, 0, stream>>>(src, dst, hB, agg, nEdges);
    sage_gemm_kernel<64, 64, 4, 64, true><<<gemmBlocks, B, 0, stream>>>(
        agg, 64, hB, 64, invd, W3l, W3r, b3, hA, 64, nTiles);

    // ---- layer 4: hA -> h4 (N x 2), no relu (weights padded to 64x16)
    zero_f32_kernel<<<blocks((long long)nNodes * 64, B), B, 0, stream>>>(agg, (long long)nNodes * 64);
    scatter_add_kernel<64, 8><<<blocks((long long)nEdges * 8, B), B, 0, stream>>>(src, dst, hA, agg, nEdges);
    sage_gemm_kernel<64, 64, 1, 2, false><<<gemmBlocks, B, 0, stream>>>(
        agg, 64, hA, 64, invd, W4lp, W4rp, b4p, h4, 2, nTiles);

    // ---- global mean pool + log_softmax
    pool_kernel<<<blocks(nNodes, B), B, 0, stream>>>(h4, batch, gsum, gcnt, nNodes);
    final_kernel<<<blocks(nGraphs, B), B, 0, stream>>>(gsum, gcnt, out, nGraphs);
}